// DefSampler_20169166422813
// MI455X (gfx1250) — compile-verified
//
#include <hip/hip_runtime.h>
#include <hip/hip_bf16.h>
#include <math.h>

typedef float v2f __attribute__((ext_vector_type(2)));
typedef float v8f __attribute__((ext_vector_type(8)));

#define NB    4
#define CIN   256
#define CCOMP 64
#define H0    64
#define W0    64
#define HW0   4096
#define H1    128
#define W1    128
#define HW1   16384

#if __has_builtin(__builtin_amdgcn_wmma_f32_16x16x4_f32)
#define HAVE_WMMA_F32 1
#else
#define HAVE_WMMA_F32 0
#endif

__device__ __forceinline__ float sigmoidf_(float t) { return 1.0f / (1.0f + expf(-t)); }

// ---------------------------------------------------------------------------
// K1: off = conv1x1(x, def_off_w,b) * sigmoid(conv1x1(x, def_ast_w,b))
// one thread per (b, oc, p), oc in [0,32)
// ---------------------------------------------------------------------------
__global__ void off_kernel(const float* __restrict__ x,
                           const float* __restrict__ ow, const float* __restrict__ ob,
                           const float* __restrict__ aw, const float* __restrict__ ab,
                           float* __restrict__ off) {
  int idx = blockIdx.x * blockDim.x + threadIdx.x;
  if (idx >= NB * 32 * HW0) return;
  int p  = idx % HW0;
  int oc = (idx / HW0) % 32;
  int b  = idx / (32 * HW0);
  const float* xp   = x + (size_t)b * CIN * HW0 + p;
  const float* wrow = ow + oc * CIN;
  const float* arow = aw + oc * CIN;
  float acc = ob[oc], accA = ab[oc];
  for (int c = 0; c < CIN; ++c) {
    float v = xp[(size_t)c * HW0];
    acc  = fmaf(v, wrow[c], acc);
    accA = fmaf(v, arow[c], accA);
  }
  off[idx] = acc * sigmoidf_(accA);
}

// ---------------------------------------------------------------------------
// K2: deformable 2x upsample (def_sample). thread per x_up element.
// ix = (ox + 0.5 + offx)/2 - 0.5 in low-res coords (derived from grid math).
// ---------------------------------------------------------------------------
__global__ void upsample_kernel(const float* __restrict__ x,
                                const float* __restrict__ off,
                                float* __restrict__ xup) {
  int idx = blockIdx.x * blockDim.x + threadIdx.x;
  if (idx >= NB * CIN * HW1) return;
  int p = idx % HW1;
  int c = (idx / HW1) % CIN;
  int b = idx / (CIN * HW1);
  int oy = p / W1, ox = p % W1;
  int g  = c >> 6;
  int hy = oy >> 1, sy = oy & 1;
  int wx = ox >> 1, sx = ox & 1;
  int ocb = g * 8 + sy * 4 + sx * 2;
  int lp  = hy * W0 + wx;
  float offx = off[((size_t)b * 32 + ocb)     * HW0 + lp];
  float offy = off[((size_t)b * 32 + ocb + 1) * HW0 + lp];
  float ix = fminf(fmaxf(((float)ox + 0.5f + offx) * 0.5f - 0.5f, 0.0f), (float)(W0 - 1));
  float iy = fminf(fmaxf(((float)oy + 0.5f + offy) * 0.5f - 0.5f, 0.0f), (float)(H0 - 1));
  float x0f = floorf(ix), y0f = floorf(iy);
  float fx = ix - x0f, fy = iy - y0f;
  int x0 = (int)x0f, y0 = (int)y0f;
  int x1 = min(x0 + 1, W0 - 1), y1 = min(y0 + 1, H0 - 1);
  const float* ch = x + ((size_t)b * CIN + c) * HW0;
  float v00 = ch[y0 * W0 + x0], v01 = ch[y0 * W0 + x1];
  float v10 = ch[y1 * W0 + x0], v11 = ch[y1 * W0 + x1];
  xup[idx] = v00 * (1.f - fx) * (1.f - fy) + v01 * fx * (1.f - fy)
           + v10 * (1.f - fx) * fy         + v11 * fx * fy;
}

// ---------------------------------------------------------------------------
// K3: comp = conv1x1(x_up, comp_w, comp_b)  -- GEMM M=NB*HW1, N=64, K=256
// One wave32 handles a 16-pixel x 64-channel tile via V_WMMA_F32_16X16X4_F32.
// A (16x4 f32) layout: lanes 0-15 -> M=lane, K={k0,k0+1}; lanes 16-31 -> same
// M, K={k0+2,k0+3}. B (4x16): lanes 0-15 -> N=lane, K={k0,k0+1}; lanes 16-31
// -> same N, K={k0+2,k0+3}. C/D: lane<16 -> N=lane, vgpr r = M=r; lane>=16 ->
// N=lane-16, vgpr r = M=8+r.
// ---------------------------------------------------------------------------
__global__ void __launch_bounds__(256) comp_wmma_kernel(const float* __restrict__ xup,
                                                        const float* __restrict__ cw,
                                                        const float* __restrict__ cb,
                                                        float* __restrict__ comp) {
  int wave = (int)((blockIdx.x * blockDim.x + threadIdx.x) >> 5);
  int lane = threadIdx.x & 31;
  int pbase = wave << 4;                 // 16 pixels per wave
  if (pbase >= NB * HW1) return;
  int b    = pbase >> 14;                // HW1 == 16384
  int ppix = pbase & (HW1 - 1);
  const float* A = xup + (size_t)b * CIN * HW1 + ppix;
  int nm = lane & 15;                    // both the A pixel row and the B/D column
#if HAVE_WMMA_F32
  int kh = (lane >> 4) << 1;             // 0 for low half, 2 for high half
  v8f acc0 = {}, acc1 = {}, acc2 = {}, acc3 = {};
  for (int k0 = 0; k0 < CIN; k0 += 4) {
    int ka = k0 + kh;
    v2f a;
    a.x = A[(size_t)ka * HW1 + nm];
    a.y = A[(size_t)(ka + 1) * HW1 + nm];
    v2f b0, b1, b2, b3;
    b0.x = cw[(0 * 16 + nm) * CIN + ka];  b0.y = cw[(0 * 16 + nm) * CIN + ka + 1];
    b1.x = cw[(1 * 16 + nm) * CIN + ka];  b1.y = cw[(1 * 16 + nm) * CIN + ka + 1];
    b2.x = cw[(2 * 16 + nm) * CIN + ka];  b2.y = cw[(2 * 16 + nm) * CIN + ka + 1];
    b3.x = cw[(3 * 16 + nm) * CIN + ka];  b3.y = cw[(3 * 16 + nm) * CIN + ka + 1];
    acc0 = __builtin_amdgcn_wmma_f32_16x16x4_f32(false, a, false, b0, (short)0, acc0, false, false);
    acc1 = __builtin_amdgcn_wmma_f32_16x16x4_f32(false, a, false, b1, (short)0, acc1, false, false);
    acc2 = __builtin_amdgcn_wmma_f32_16x16x4_f32(false, a, false, b2, (short)0, acc2, false, false);
    acc3 = __builtin_amdgcn_wmma_f32_16x16x4_f32(false, a, false, b3, (short)0, acc3, false, false);
  }
  int m0 = (lane >> 4) << 3;             // M = r (low half) or 8+r (high half)
  float bias0 = cb[0 * 16 + nm], bias1 = cb[1 * 16 + nm];
  float bias2 = cb[2 * 16 + nm], bias3 = cb[3 * 16 + nm];
#pragma unroll
  for (int r = 0; r < 8; ++r) {
    int mm = m0 + r;
    comp[((size_t)b * CCOMP + 0 * 16 + nm) * HW1 + ppix + mm] = acc0[r] + bias0;
    comp[((size_t)b * CCOMP + 1 * 16 + nm) * HW1 + ppix + mm] = acc1[r] + bias1;
    comp[((size_t)b * CCOMP + 2 * 16 + nm) * HW1 + ppix + mm] = acc2[r] + bias2;
    comp[((size_t)b * CCOMP + 3 * 16 + nm) * HW1 + ppix + mm] = acc3[r] + bias3;
  }
#else
  // Scalar fallback: lane handles pixel nm, 32 output channels per lane half.
  int obase = (lane >> 4) * 32;
  for (int o = 0; o < 32; ++o) {
    int oo = obase + o;
    float acc = cb[oo];
    for (int k = 0; k < CIN; ++k)
      acc = fmaf(A[(size_t)k * HW1 + nm], cw[oo * CIN + k], acc);
    comp[((size_t)b * CCOMP + oo) * HW1 + ppix + nm] = acc;
  }
#endif
}

// ---------------------------------------------------------------------------
// K4: kern = softmax(conv1x1(comp, filt_w, filt_b)) over the 9 taps.
// one thread per (b, p); 9 accumulators over 64 comp channels.
// ---------------------------------------------------------------------------
__global__ void kern_kernel(const float* __restrict__ comp,
                            const float* __restrict__ fw, const float* __restrict__ fb,
                            float* __restrict__ kern) {
  int idx = blockIdx.x * blockDim.x + threadIdx.x;
  if (idx >= NB * HW1) return;
  int p = idx % HW1;
  int b = idx / HW1;
  float l[9];
#pragma unroll
  for (int k = 0; k < 9; ++k) l[k] = fb[k];
  const float* cp = comp + (size_t)b * CCOMP * HW1 + p;
  for (int c = 0; c < CCOMP; ++c) {
    float v = cp[(size_t)c * HW1];
#pragma unroll
    for (int k = 0; k < 9; ++k) l[k] = fmaf(v, fw[k * CCOMP + c], l[k]);
  }
  float mx = l[0];
#pragma unroll
  for (int k = 1; k < 9; ++k) mx = fmaxf(mx, l[k]);
  float s = 0.f;
#pragma unroll
  for (int k = 0; k < 9; ++k) { l[k] = expf(l[k] - mx); s += l[k]; }
  float inv = 1.0f / s;
#pragma unroll
  for (int k = 0; k < 9; ++k)
    kern[((size_t)b * 9 + k) * HW1 + p] = l[k] * inv;
}

// ---------------------------------------------------------------------------
// K5: x_filt = sum_k kern[k] * shifted(x_up)   (3x3, zero pad)
// ---------------------------------------------------------------------------
__global__ void filt_kernel(const float* __restrict__ xup,
                            const float* __restrict__ kern,
                            float* __restrict__ xf) {
  int idx = blockIdx.x * blockDim.x + threadIdx.x;
  if (idx >= NB * CIN * HW1) return;
  int p = idx % HW1;
  int c = (idx / HW1) % CIN;
  int b = idx / (CIN * HW1);
  int y = p / W1, x = p % W1;
  const float* ch = xup + ((size_t)b * CIN + c) * HW1;
  const float* kp = kern + (size_t)b * 9 * HW1 + p;
  float acc = 0.f;
#pragma unroll
  for (int dy = 0; dy < 3; ++dy) {
#pragma unroll
    for (int dx = 0; dx < 3; ++dx) {
      int yy = y + dy - 1, xx = x + dx - 1;
      float v = (yy >= 0 && yy < H1 && xx >= 0 && xx < W1) ? ch[yy * W1 + xx] : 0.f;
      acc = fmaf(v, kp[(size_t)(dy * 3 + dx) * HW1], acc);
    }
  }
  xf[idx] = acc;
}

// ---------------------------------------------------------------------------
// K6: toff = conv1x1(comp, trim_w,b) * sigmoid(conv1x1(comp, trim_ast_w,b))
// one thread per (b, j, p), j in [0,8)
// ---------------------------------------------------------------------------
__global__ void toff_kernel(const float* __restrict__ comp,
                            const float* __restrict__ tw, const float* __restrict__ tb,
                            const float* __restrict__ taw, const float* __restrict__ tab,
                            float* __restrict__ toff) {
  int idx = blockIdx.x * blockDim.x + threadIdx.x;
  if (idx >= NB * 8 * HW1) return;
  int p = idx % HW1;
  int j = (idx / HW1) % 8;
  int b = idx / (8 * HW1);
  const float* cp = comp + (size_t)b * CCOMP * HW1 + p;
  float acc = tb[j], accA = tab[j];
  for (int c = 0; c < CCOMP; ++c) {
    float v = cp[(size_t)c * HW1];
    acc  = fmaf(v, tw[j * CCOMP + c], acc);
    accA = fmaf(v, taw[j * CCOMP + c], accA);
  }
  toff[idx] = acc * sigmoidf_(accA);
}

// ---------------------------------------------------------------------------
// K7: out = trim_op(x_filt, toff): ix = x + offx (clamped), bilinear gather.
// Non-temporal final store: out is write-once, keep L2 for intermediates.
// ---------------------------------------------------------------------------
__global__ void trim_kernel(const float* __restrict__ xf,
                            const float* __restrict__ toff,
                            float* __restrict__ out) {
  int idx = blockIdx.x * blockDim.x + threadIdx.x;
  if (idx >= NB * CIN * HW1) return;
  int p = idx % HW1;
  int c = (idx / HW1) % CIN;
  int b = idx / (CIN * HW1);
  int y = p / W1, x = p % W1;
  int g = c >> 6;
  float offx = toff[((size_t)b * 8 + g * 2)     * HW1 + p];
  float offy = toff[((size_t)b * 8 + g * 2 + 1) * HW1 + p];
  float ix = fminf(fmaxf((float)x + offx, 0.0f), (float)(W1 - 1));
  float iy = fminf(fmaxf((float)y + offy, 0.0f), (float)(H1 - 1));
  float x0f = floorf(ix), y0f = floorf(iy);
  float fx = ix - x0f, fy = iy - y0f;
  int x0 = (int)x0f, y0 = (int)y0f;
  int x1 = min(x0 + 1, W1 - 1), y1 = min(y0 + 1, H1 - 1);
  const float* ch = xf + ((size_t)b * CIN + c) * HW1;
  float v00 = ch[y0 * W1 + x0], v01 = ch[y0 * W1 + x1];
  float v10 = ch[y1 * W1 + x0], v11 = ch[y1 * W1 + x1];
  float r = v00 * (1.f - fx) * (1.f - fy) + v01 * fx * (1.f - fy)
          + v10 * (1.f - fx) * fy         + v11 * fx * fy;
  __builtin_nontemporal_store(r, &out[idx]);
}

// ---------------------------------------------------------------------------
extern "C" void kernel_launch(void* const* d_in, const int* in_sizes, int n_in,
                              void* d_out, int out_size, void* d_ws, size_t ws_size,
                              hipStream_t stream) {
  const float* x    = (const float*)d_in[0];
  const float* dow  = (const float*)d_in[1];
  const float* dob  = (const float*)d_in[2];
  const float* daw  = (const float*)d_in[3];
  const float* dab  = (const float*)d_in[4];
  const float* cw   = (const float*)d_in[5];
  const float* cbp  = (const float*)d_in[6];
  const float* fw   = (const float*)d_in[7];
  const float* fb   = (const float*)d_in[8];
  const float* tw   = (const float*)d_in[9];
  const float* tb   = (const float*)d_in[10];
  const float* taw  = (const float*)d_in[11];
  const float* tab  = (const float*)d_in[12];
  float* out = (float*)d_out;

  // workspace partition (floats) -- ~157 MB total, L2-resident on MI455X
  float* ws    = (float*)d_ws;
  float* off   = ws;                               // NB*32*HW0      =   524288
  float* xup   = off  + (size_t)NB * 32 * HW0;     // NB*CIN*HW1     = 16777216
  float* comp  = xup  + (size_t)NB * CIN * HW1;    // NB*64*HW1      =  4194304
  float* kern  = comp + (size_t)NB * CCOMP * HW1;  // NB*9*HW1       =   589824
  float* toff  = kern + (size_t)NB * 9 * HW1;      // NB*8*HW1       =   524288
  float* xf    = toff + (size_t)NB * 8 * HW1;      // NB*CIN*HW1     = 16777216
  (void)ws_size; (void)n_in; (void)in_sizes; (void)out_size;

  const int T = 256;
  int n1 = NB * 32 * HW0;
  off_kernel<<<(n1 + T - 1) / T, T, 0, stream>>>(x, dow, dob, daw, dab, off);

  int n2 = NB * CIN * HW1;
  upsample_kernel<<<(n2 + T - 1) / T, T, 0, stream>>>(x, off, xup);

  int waves = (NB * HW1) / 16;                      // 16 pixels per wave32
  comp_wmma_kernel<<<(waves * 32 + T - 1) / T, T, 0, stream>>>(xup, cw, cbp, comp);

  int n4 = NB * HW1;
  kern_kernel<<<(n4 + T - 1) / T, T, 0, stream>>>(comp, fw, fb, kern);

  filt_kernel<<<(n2 + T - 1) / T, T, 0, stream>>>(xup, kern, xf);

  int n6 = NB * 8 * HW1;
  toff_kernel<<<(n6 + T - 1) / T, T, 0, stream>>>(comp, tw, tb, taw, tab, toff);

  trim_kernel<<<(n2 + T - 1) / T, T, 0, stream>>>(xf, toff, out);
}